// Deep_Mem_RelativeLocs_ProjectedLowerDim_40089224741407
// MI455X (gfx1250) — compile-verified
//
#include <hip/hip_runtime.h>

typedef __attribute__((ext_vector_type(2))) float v2f;
typedef __attribute__((ext_vector_type(8))) float v8f;
typedef __attribute__((ext_vector_type(4))) float f4v;
// 4-byte-aligned v2f for global loads (rows are only dword-aligned: 241 is odd).
// AMDGPU supports dword-aligned b64 global loads, so this still merges.
typedef v2f u_v2f __attribute__((aligned(4)));

#define K_DIM    241
#define K_STRIDE 246       // LDS row stride in dwords: even (b64-aligned), gcd(246,64)=2
#define N_COLS   7
#define NPAD     16
#define NFLAT    80000
#define OUT_N4   4000000   // 16,000,000 floats / 4

// ---------------------------------------------------------------- zero bins
__global__ void zero_counts_kernel(float* __restrict__ counts) {
    int i = blockIdx.x * blockDim.x + threadIdx.x;
    if (i < NFLAT) counts[i] = 0.0f;
}

// ------------------------------------------------- WMMA GEMM + hash + hist
// One wave computes a 16-row x 16-col fp32 tile of rel_vec @ W^T using
// V_WMMA_F32_16X16X4_F32, then hashes its 16 rows into the histogram.
__global__ __launch_bounds__(256) void hash_hist_kernel(
        const float* __restrict__ rel,     // [n_rows][241]
        const float* __restrict__ hw,      // [7][241]
        float* __restrict__ counts,        // [80000]
        int n_rows) {
    __shared__ float Bl[NPAD * K_STRIDE];  // B transposed: Bl[n][k], zero-padded
    __shared__ float tile[8][16][17];      // per-wave D tile (17 = bank-pad)

    const int tid = threadIdx.x;

    // Stage B: Bl[n*K_STRIDE + k] = hw[n*241 + k]; zeros for k>=241 or n>=7.
    for (int idx = tid; idx < NPAD * K_STRIDE; idx += blockDim.x) {
        int n = idx / K_STRIDE;
        int k = idx % K_STRIDE;
        float v = 0.0f;
        if (n < N_COLS && k < K_DIM) v = hw[n * K_DIM + k];
        Bl[idx] = v;
    }
    __syncthreads();

    const int wave   = tid >> 5;           // 8 waves / block (wave32)
    const int lane   = tid & 31;
    const int laneM  = lane & 15;          // M (A rows) / N (B cols)
    const int laneHi = lane >> 4;          // K-pair selector (0 or 1)

    const long long r0  = ((long long)blockIdx.x * 8 + wave) * 16;
    long long       row = r0 + laneM;
    if (row > n_rows - 1) row = n_rows - 1;          // clamp (no OOB reads)

    // Fragment element j has K = k0 + 2*laneHi + j  ->  both A floats and both
    // B floats are contiguous: one b64 each.
    const float* __restrict__ ap = rel + row * (long long)K_DIM + 2 * laneHi;
    const float* __restrict__ bp = Bl + laneM * K_STRIDE + 2 * laneHi;

    v8f c0 = {};
    v8f c1 = {};

    // Main loop: k = 0..239, no bounds checks (max K touched = 236+2+1 = 239).
    #pragma unroll 4
    for (int k0 = 0; k0 < 240; k0 += 4) {
        if ((k0 & 15) == 0)
            __builtin_prefetch(ap + k0 + 256, 0, 1);   // global_prefetch_b8
        v2f a = *(const u_v2f*)(ap + k0);              // global_load_b64
        v2f b = *(const v2f*)(bp + k0);                // ds_load_b64 (8B aligned)
        if (k0 & 4) {
            c1 = __builtin_amdgcn_wmma_f32_16x16x4_f32(
                    false, a, false, b, (short)0, c1, false, false);
        } else {
            c0 = __builtin_amdgcn_wmma_f32_16x16x4_f32(
                    false, a, false, b, (short)0, c0, false, false);
        }
    }

    // Tail step: k = 240..243; only k=240 is real. B is zero-padded there,
    // and we zero the A lanes whose K >= 241 to avoid OOB/NaN propagation.
    {
        v2f a, b;
        a.x = (laneHi == 0) ? ap[240] : 0.0f;   // K = 240 + 2*laneHi
        a.y = 0.0f;                             // K = 241 + 2*laneHi, always pad
        b   = *(const v2f*)(bp + 240);          // zeros beyond 240 by staging
        c0 = __builtin_amdgcn_wmma_f32_16x16x4_f32(
                false, a, false, b, (short)0, c0, false, false);
    }

    // Spill D tile: element (lane, vgpr j) is (M = j + 8*laneHi, N = laneM).
    #pragma unroll
    for (int j = 0; j < 8; ++j) {
        tile[wave][j + 8 * laneHi][laneM] = c0[j] + c1[j];
    }
    __syncthreads();

    // Lanes 0..15: hash one row each, scatter-add into histogram.
    if (lane < 16) {
        const long long r = r0 + lane;
        if (r < n_rows) {
            // IDX_DIMS = (2,10,10,2,10,10,2); clip(h,0,19) then min(dim-1)
            // collapses to clamp(h, 0, dim-1) since dim-1 <= 9 < 19.
            const int dm1[7] = {1, 9, 9, 1, 9, 9, 1};
            const int str[7] = {40000, 4000, 400, 200, 20, 2, 1};
            int flat = 0;
            #pragma unroll
            for (int n = 0; n < 7; ++n) {
                int h = (int)tile[wave][lane][n];   // trunc, same as astype(int32)
                h = h < 0 ? 0 : (h > dm1[n] ? dm1[n] : h);
                flat += h * str[n];
            }
            atomicAdd(&counts[flat], 1.0f);         // global_atomic_add_f32
        }
    }
}

// ------------------------------------------------- broadcast counts -> out
// out[i] = mem[i] + counts[i/200], float4-vectorized (200 % 4 == 0).
__global__ void finalize_kernel(const float* __restrict__ mem,
                                const float* __restrict__ counts,
                                float* __restrict__ out) {
    long long i = (long long)blockIdx.x * blockDim.x + threadIdx.x;
    if (i < OUT_N4) {
        f4v v = ((const f4v*)mem)[i];
        float cv = counts[(int)((i * 4) / 200)];
        v.x += cv; v.y += cv; v.z += cv; v.w += cv;
        ((f4v*)out)[i] = v;
    }
}

// ---------------------------------------------------------------- launcher
extern "C" void kernel_launch(void* const* d_in, const int* in_sizes, int n_in,
                              void* d_out, int out_size, void* d_ws, size_t ws_size,
                              hipStream_t stream) {
    const float* rel = (const float*)d_in[0];   // rel_vec  [n_rows*241]
    const float* hw  = (const float*)d_in[1];   // hash_w   [7*241]
    const float* mem = (const float*)d_in[2];   // mem      [16,000,000]
    float* out    = (float*)d_out;
    float* counts = (float*)d_ws;               // 80000 floats = 320 KB scratch

    const int n_rows = in_sizes[0] / K_DIM;     // 415744

    zero_counts_kernel<<<(NFLAT + 255) / 256, 256, 0, stream>>>(counts);

    const int rows_per_block = 8 * 16;          // 8 waves x 16 rows
    const int gemm_blocks = (n_rows + rows_per_block - 1) / rows_per_block;
    hash_hist_kernel<<<gemm_blocks, 256, 0, stream>>>(rel, hw, counts, n_rows);

    finalize_kernel<<<(OUT_N4 + 255) / 256, 256, 0, stream>>>(mem, counts, out);
}